// SentimentCapsule_17540646437027
// MI455X (gfx1250) — compile-verified
//
#include <hip/hip_runtime.h>

typedef float v2f __attribute__((ext_vector_type(2)));
typedef float v8f __attribute__((ext_vector_type(8)));

constexpr int NBATCH  = 4;            // batch items per workgroup
constexpr int K2      = 100;          // capsules (rows of t per batch)
constexpr int A2      = 25;           // inner / output dim
constexpr int TSTRIDE = 32;           // padded column stride of t in LDS
constexpr int ROWS    = NBATCH * K2;  // 400 rows of t per workgroup
constexpr int NTILES  = ROWS / 16;    // 25 WMMA M-tiles
constexpr int ITERS   = 3;
constexpr int KQ      = 7;            // K padded to 28 = 7 quads of 4

__global__ __launch_bounds__(256) void capsule_route_kernel(
    const float* __restrict__ logic,   // (B,100,25) row-major
    const float* __restrict__ Wcap,    // (25,25) row-major, t = logic @ W^T
    float* __restrict__ out)           // (B,100)
{
  __shared__ float t_lds[ROWS * TSTRIDE];   // 51200 B
  __shared__ float c_lds[NBATCH * 32];      //   512 B
  __shared__ float o_lds[NBATCH * 128];     //  2048 B

  const int tid  = threadIdx.x;
  // wave index is wave-uniform; readfirstlane lets the compiler keep the
  // tile loop + phase-2 select in pure SALU control flow (no exec juggling).
  const int wave = __builtin_amdgcn_readfirstlane(tid >> 5);
  const int lane = tid & 31;
  const int lo   = lane & 15;     // M (or N) index within tile
  const int hi   = lane >> 4;     // half select
  const int kh   = hi << 1;       // K offset within each K=4 quad

  // ---- B operand: B[a][x] = Wcap[x][a], zero-padded to 28x32. ----
  // V_WMMA_F32_16X16X4_F32 B layout (mirror of A table): VGPR0 holds K=kh,
  // VGPR1 holds K=kh+1, N = lane&15 in both halves.
  float bw0[KQ][2], bw1[KQ][2];
  #pragma unroll
  for (int j = 0; j < KQ; ++j) {
    const int a0 = 4 * j + kh, a1 = a0 + 1;
    {
      const int x = lo;                       // N-tile 0: x in [0,16) < 25
      bw0[j][0] = (a0 < A2) ? Wcap[x * A2 + a0] : 0.0f;
      bw0[j][1] = (a1 < A2) ? Wcap[x * A2 + a1] : 0.0f;
    }
    {
      const int x = 16 + lo;                  // N-tile 1: x in [16,32)
      bw1[j][0] = (x < A2 && a0 < A2) ? Wcap[x * A2 + a0] : 0.0f;
      bw1[j][1] = (x < A2 && a1 < A2) ? Wcap[x * A2 + a1] : 0.0f;
    }
  }

  const long long wgRow0 = (long long)blockIdx.x * ROWS;

  // ---- Phase 1: t = logic @ W^T via chained V_WMMA_F32_16X16X4_F32 ----
  for (int tile = wave; tile < NTILES; tile += 8) {
    const float* rowp = logic + (wgRow0 + tile * 16 + lo) * A2;
    // Unconditional speculative prefetch of the next tile this wave will do;
    // out-of-range speculative prefetches are silently dropped.
    __builtin_prefetch(rowp + 8 * 16 * A2, 0, 1);

    // A layout (ISA 16x4 f32): lane&15 = M, halves carry K={kh,kh+1}.
    float av[KQ][2];
    #pragma unroll
    for (int j = 0; j < KQ; ++j) {
      const int e0 = 4 * j + kh, e1 = e0 + 1;
      av[j][0] = (e0 < A2) ? rowp[e0] : 0.0f;
      av[j][1] = (e1 < A2) ? rowp[e1] : 0.0f;
    }

    v8f acc0 = {};
    v8f acc1 = {};
    #pragma unroll
    for (int j = 0; j < KQ; ++j) {
      v2f a;  a.x  = av[j][0];  a.y  = av[j][1];
      v2f b0; b0.x = bw0[j][0]; b0.y = bw0[j][1];
      v2f b1; b1.x = bw1[j][0]; b1.y = bw1[j][1];
      acc0 = __builtin_amdgcn_wmma_f32_16x16x4_f32(
          false, a, false, b0, (short)0, acc0, false, false);
      acc1 = __builtin_amdgcn_wmma_f32_16x16x4_f32(
          false, a, false, b1, (short)0, acc1, false, false);
    }

    // C layout: VGPR v -> M = v + 8*hi, N = lo. Padding cols are exact zeros.
    #pragma unroll
    for (int v = 0; v < 8; ++v) {
      const int lr = tile * 16 + v + 8 * hi;   // local row = blocal*100 + k
      t_lds[lr * TSTRIDE + lo]      = acc0[v];
      t_lds[lr * TSTRIDE + 16 + lo] = acc1[v];
    }
  }
  __syncthreads();

  // ---- Phase 2: dynamic routing, one wave per batch item ----
  if (wave < NBATCH) {
    float* tb = &t_lds[wave * K2 * TSTRIDE];
    float* cb = &c_lds[wave * 32];
    float* ob = &o_lds[wave * 128];
    const int a = lane;             // this lane owns b[a], a < 25
    float bacc = 0.0f;
    float ov[4] = {0.0f, 0.0f, 0.0f, 0.0f};

    for (int it = 0; it < ITERS; ++it) {
      // c = softmax(b) over 25 entries (lanes >= 25 masked out)
      float bv = (a < A2) ? bacc : -__builtin_inff();
      float mx = bv;
      #pragma unroll
      for (int off = 16; off > 0; off >>= 1)
        mx = fmaxf(mx, __shfl_xor(mx, off, 32));
      float ex = (a < A2) ? __expf(bacc - mx) : 0.0f;
      float sm = ex;
      #pragma unroll
      for (int off = 16; off > 0; off >>= 1)
        sm += __shfl_xor(sm, off, 32);
      cb[lane] = ex / sm;                       // 0 for lanes >= 25

      float cr[A2];
      #pragma unroll
      for (int i = 0; i < A2; ++i) cr[i] = cb[i];

      // s[k] = t[k,:] . c ; lanes parallel over k; partial ||s||^2
      float n2p = 0.0f;
      #pragma unroll
      for (int i = 0; i < 4; ++i) {
        const int k = lane + 32 * i;
        float s = 0.0f;
        if (k < K2) {
          const float* tr = &tb[k * TSTRIDE];
          #pragma unroll
          for (int aa = 0; aa < A2; ++aa) s = fmaf(tr[aa], cr[aa], s);
          n2p += s * s;
        }
        ov[i] = s;
      }
      float n2 = n2p;
      #pragma unroll
      for (int off = 16; off > 0; off >>= 1)
        n2 += __shfl_xor(n2, off, 32);

      // squash: o = (n2/(1+n2)) * s/|s| = s * sqrt(n2)/(1+n2)
      const float scale = sqrtf(n2) / (1.0f + n2);
      #pragma unroll
      for (int i = 0; i < 4; ++i) {
        const int k = lane + 32 * i;
        ov[i] *= scale;
        if (k < K2) ob[k] = ov[i];
      }

      // b += t^T o  (not needed after the last iteration)
      if (it != ITERS - 1 && a < A2) {
        float bu = 0.0f;
        for (int k = 0; k < K2; ++k)
          bu = fmaf(ob[k], tb[k * TSTRIDE + a], bu);
        bacc += bu;
      }
    }

    const long long obase =
        ((long long)blockIdx.x * NBATCH + wave) * (long long)K2;
    #pragma unroll
    for (int i = 0; i < 4; ++i) {
      const int k = lane + 32 * i;
      if (k < K2) out[obase + k] = ov[i];
    }
  }
}

extern "C" void kernel_launch(void* const* d_in, const int* in_sizes, int n_in,
                              void* d_out, int out_size, void* d_ws, size_t ws_size,
                              hipStream_t stream) {
  const float* logic = (const float*)d_in[0];   // (65536,100,25) f32
  const float* Wcap  = (const float*)d_in[1];   // (25,25) f32
  float* out = (float*)d_out;                   // (65536,100) f32

  const int B = in_sizes[0] / (K2 * A2);        // 65536
  dim3 grid(B / NBATCH), block(256);
  capsule_route_kernel<<<grid, block, 0, stream>>>(logic, Wcap, out);
}